// my_GCN_51445118271508
// MI455X (gfx1250) — compile-verified
//
#include <hip/hip_runtime.h>
#include <math.h>

typedef __attribute__((ext_vector_type(2))) float v2f;
typedef __attribute__((ext_vector_type(8))) float v8f;

#define N_NODES 2500
#define D_FEAT 2
#define WAVES_PER_BLOCK 4
#define ROWS_PER_WAVE 16
#define ROWS_PER_BLOCK (WAVES_PER_BLOCK * ROWS_PER_WAVE)  // 64
#define BCOL_P 2504  // padded plane stride (multiple of 8 floats)

// One GCN layer: h' = h + relu(adj @ (h W^T + b))
// grid = (ceil(n/64), B), block = 128 threads (4 waves of 32).
__global__ __launch_bounds__(128) void gcn_layer_kernel(
    const float* __restrict__ hsrc,   // [B, n, 2]
    const float* __restrict__ adj,    // [B, n, n]
    const float* __restrict__ W,      // [2, 2] row-major (e, d)
    const float* __restrict__ bias,   // [2]
    float* __restrict__ hdst,         // [B, n, 2]
    int n)                            // runtime to keep loops as loops
{
  __shared__ __align__(16) float h_s[N_NODES * 2];
  // pre, column-major, plus an always-zero third plane for WMMA cols 2..15.
  __shared__ __align__(16) float bcol[3 * BCOL_P];

  const int b   = blockIdx.y;
  const int tid = threadIdx.x;

  // Uniform layer params (scalar loads)
  const float w00 = W[0], w01 = W[1], w10 = W[2], w11 = W[3];
  const float bb0 = bias[0], bb1 = bias[1];

  // Phase 1: stage h[b] into LDS; pre = h @ W^T + b, column-major; zero plane 2.
  const float* hb = hsrc + (size_t)b * n * 2;
  for (int j = tid; j < n; j += 128) {
    v2f h = *(const v2f*)(hb + j * 2);
    ((v2f*)h_s)[j] = h;
    bcol[j]                = fmaf(h.x, w00, fmaf(h.y, w01, bb0));
    bcol[BCOL_P + j]       = fmaf(h.x, w10, fmaf(h.y, w11, bb1));
    bcol[2 * BCOL_P + j]   = 0.0f;
  }
  __syncthreads();

  const int lane  = tid & 31;
  const int col   = lane & 15;   // output column; real data iff col < 2
  const int khalf = lane >> 4;   // 0 -> K=k,k+1 ; 1 -> K=k+2,k+3 (32-bit layout)

  const int rowBase = (blockIdx.x * WAVES_PER_BLOCK + (tid >> 5)) * ROWS_PER_WAVE;
  if (rowBase < n) {             // wave-uniform: EXEC all-ones inside
    const int  row      = rowBase + col;           // A-matrix row for this lane
    const bool rowValid = (row < n);

    const float* aptr = adj + (size_t)b * n * n
                            + (size_t)(rowValid ? row : 0) * n + 2 * khalf;
    // B operand base: lanes col>=2 point at the zero plane. No per-iter predication.
    const int bbase = (col < 2 ? col : 2) * BCOL_P + 2 * khalf;

    v8f acc = {0.f, 0.f, 0.f, 0.f, 0.f, 0.f, 0.f, 0.f};

    if (rowBase + ROWS_PER_WAVE <= n) {
      // Fast path: whole 16-row tile valid (156 of 157 tiles).
      #pragma unroll 4
      for (int k = 0; k < n; k += 4) {
        v2f a  = *(const v2f*)(aptr + k);            // global_load_b64
        v2f bm = *(const v2f*)(&bcol[bbase + k]);    // ds_load_b64
        acc = __builtin_amdgcn_wmma_f32_16x16x4_f32(
            false, a, false, bm, (short)0, acc, false, false);
      }
    } else {
      // Edge tile: zero A rows beyond n (address already clamped to row 0).
      #pragma unroll 2
      for (int k = 0; k < n; k += 4) {
        v2f a  = *(const v2f*)(aptr + k);
        if (!rowValid) { a.x = 0.f; a.y = 0.f; }
        v2f bm = *(const v2f*)(&bcol[bbase + k]);
        acc = __builtin_amdgcn_wmma_f32_16x16x4_f32(
            false, a, false, bm, (short)0, acc, false, false);
      }
    }

    // Phase 3: residual + relu, scatter. C/D layout: row = v + 8*khalf, col = lane&15.
    if (col < 2) {
      #pragma unroll
      for (int v = 0; v < 8; ++v) {
        const int r = rowBase + v + khalf * 8;
        if (r < n) {
          const float agg = acc[v];
          const float val = h_s[r * 2 + col] + (agg > 0.f ? agg : 0.f);
          hdst[((size_t)b * n + r) * 2 + col] = val;
        }
      }
    }
  }
}

// out = sigmoid(h @ Wout^T + bout), one thread per (b, n) node.
__global__ __launch_bounds__(256) void gcn_out_kernel(
    const float* __restrict__ h, const float* __restrict__ Wout,
    const float* __restrict__ bout, float* __restrict__ out, int total)
{
  const int i = blockIdx.x * blockDim.x + threadIdx.x;
  if (i >= total) return;
  const v2f hv = ((const v2f*)h)[i];
  const float o0 = fmaf(hv.x, Wout[0], fmaf(hv.y, Wout[1], bout[0]));
  const float o1 = fmaf(hv.x, Wout[2], fmaf(hv.y, Wout[3], bout[1]));
  v2f r;
  r.x = 1.0f / (1.0f + __expf(-o0));
  r.y = 1.0f / (1.0f + __expf(-o1));
  ((v2f*)out)[i] = r;
}

extern "C" void kernel_launch(void* const* d_in, const int* in_sizes, int n_in,
                              void* d_out, int out_size, void* d_ws, size_t ws_size,
                              hipStream_t stream) {
  const float* x    = (const float*)d_in[0];  // [16, 2500, 2]
  const float* adj  = (const float*)d_in[1];  // [16, 2500, 2500]
  const float* Ws   = (const float*)d_in[2];  // [25, 2, 2]
  const float* bs   = (const float*)d_in[3];  // [25, 2]
  const float* Wout = (const float*)d_in[4];  // [2, 2]
  const float* bout = (const float*)d_in[5];  // [2]

  const int B = 16, D = D_FEAT, L = 25, n = N_NODES;
  const size_t hElems = (size_t)B * n * D;    // 80000
  float* bufA = (float*)d_ws;
  float* bufB = bufA + hElems;

  dim3 grid((n + ROWS_PER_BLOCK - 1) / ROWS_PER_BLOCK, B);  // (40, 16)
  for (int l = 0; l < L; ++l) {
    const float* src = (l == 0) ? x : (((l - 1) & 1) ? bufB : bufA);
    float* dst = (l & 1) ? bufB : bufA;
    gcn_layer_kernel<<<grid, 128, 0, stream>>>(src, adj, Ws + l * D * D,
                                               bs + l * D, dst, n);
  }
  // 25 layers: last layer index 24 (even) wrote bufA.
  const int total = B * n;
  gcn_out_kernel<<<(total + 255) / 256, 256, 0, stream>>>(bufA, Wout, bout,
                                                          (float*)d_out, total);
}